// RTGCNGraphConvolution_37864431682671
// MI455X (gfx1250) — compile-verified
//
#include <hip/hip_runtime.h>
#include <hip/hip_bf16.h>

#define NN 8192   // num nodes
#define HH 16     // hidden units
#define BB 4      // batch

typedef __attribute__((ext_vector_type(2))) float v2f;
typedef __attribute__((ext_vector_type(8))) float v8f;

// ---------------------------------------------------------------------------
// Stage 1: per-node gating.
//   cm = relu([x*iw_m, h_m] @ Wm + bm);  cv = relu([x*iw_v, h_v] @ Wv + bv)
//   nw = exp(-cv)
//   M0[m, b*16+h] = cm*nw        (stored k-paired for WMMA B-operand loads)
//   M1[m, b*16+h] = cv*nw*nw
// Paired layout: idx = (m>>1)*128 + col*2 + (m&1), col = b*16+h.
// ---------------------------------------------------------------------------
__global__ __launch_bounds__(256) void rtgcn_stage1(
    const float* __restrict__ inputs,   // (B, N)
    const float* __restrict__ hidden,   // (B, 2*N*H)
    const float* __restrict__ Wm,       // (17, 16)
    const float* __restrict__ bm,       // (16)
    const float* __restrict__ Wv,       // (17, 16)
    const float* __restrict__ bv,       // (16)
    const float* __restrict__ iwm,      // (1)
    const float* __restrict__ iwv,      // (1)
    float* __restrict__ M0,             // (N/2, 128) paired
    float* __restrict__ M1)
{
    __shared__ float sWm[17 * 16];
    __shared__ float sWv[17 * 16];
    __shared__ float sb[32];

    const int t = threadIdx.x;
    if (t < 272) { sWm[t] = Wm[t]; sWv[t] = Wv[t]; }
    if (t < 16)  { sb[t] = bm[t]; sb[16 + t] = bv[t]; }
    __syncthreads();

    const int gid = blockIdx.x * 256 + t;      // 0 .. B*N-1  (grid sized exactly)
    const int b = gid >> 13;                   // gid / N
    const int m = gid & (NN - 1);              // gid % N

    const float x  = inputs[b * NN + m];
    const float xm = x * iwm[0];
    const float xv = x * iwv[0];

    const float* hm = hidden + (size_t)b * (2 * NN * HH) + (size_t)m * HH;
    const float* hv = hm + (size_t)NN * HH;

    float hmv[16], hvv[16];
#pragma unroll
    for (int j = 0; j < 16; ++j) { hmv[j] = hm[j]; hvv[j] = hv[j]; }

    const size_t base = (size_t)(m >> 1) * 128 + (size_t)(b * 16) * 2 + (size_t)(m & 1);

#pragma unroll
    for (int h = 0; h < 16; ++h) {
        float cm = sb[h]      + xm * sWm[h];          // Wm[0, h]
        float cv = sb[16 + h] + xv * sWv[h];          // Wv[0, h]
#pragma unroll
        for (int j = 0; j < 16; ++j) {
            cm = fmaf(hmv[j], sWm[(1 + j) * 16 + h], cm);
            cv = fmaf(hvv[j], sWv[(1 + j) * 16 + h], cv);
        }
        cm = fmaxf(cm, 0.0f);
        cv = fmaxf(cv, 0.0f);
        const float nw = __expf(-cv);                 // LAMDA = 1
        M0[base + 2 * h] = cm * nw;
        M1[base + 2 * h] = cv * nw * nw;
    }
}

// ---------------------------------------------------------------------------
// Stage 2: out[b, n, c] = sum_m lap[n, m] * M[m, b*16 + (c mod 16)]
// One wave per (16-row tile, laplacian). 4 batch accumulators so each
// Laplacian tile streams from HBM exactly once. V_WMMA_F32_16X16X4_F32,
// K-step of 4 over K = 8192.
// ---------------------------------------------------------------------------
__global__ __launch_bounds__(32) void rtgcn_stage2(
    const float* __restrict__ lap0,
    const float* __restrict__ lap1,
    const float* __restrict__ M0,      // paired (N/2, 128)
    const float* __restrict__ M1,
    float* __restrict__ out)           // (B, N, 32): [:,:,0:16]=mean, [16:32]=var
{
    const int lane = threadIdx.x;      // 0..31
    const int half = lane >> 4;        // 0: K = k,k+1   1: K = k+2,k+3
    const int l16  = lane & 15;
    const int n0   = blockIdx.x * 16;  // output row tile
    const int which = blockIdx.y;      // 0 = mean path (lap0), 1 = var path (lap1)

    const float* __restrict__ lap = which ? lap1 : lap0;
    const float* __restrict__ M   = which ? M1   : M0;

    // A-operand stream: this lane reads lap[n0 + l16, k + 2*half .. +1]
    const float* aptr = lap + (size_t)(n0 + l16) * NN + (size_t)(2 * half);
    // B-operand stream (paired layout): pair index = (k>>1) + half, col = b*16 + l16
    const float* bptr = M + (size_t)half * 128 + (size_t)l16 * 2;

    v8f acc[4] = {};   // one 16x16 f32 accumulator per batch

    for (int ko = 0; ko < NN; ko += 64) {
        if (ko + 1024 < NN)
            __builtin_prefetch(aptr + ko + 1024, 0, 3);   // global_prefetch_b8, 4KB ahead
#pragma unroll 4
        for (int k = ko; k < ko + 64; k += 4) {
            v2f a = *(const v2f*)(aptr + k);
            const float* bk = bptr + (size_t)(k >> 1) * 128;
            v2f b0 = *(const v2f*)(bk);        // batch 0, cols 0..15
            v2f b1 = *(const v2f*)(bk + 32);   // batch 1
            v2f b2 = *(const v2f*)(bk + 64);   // batch 2
            v2f b3 = *(const v2f*)(bk + 96);   // batch 3
            acc[0] = __builtin_amdgcn_wmma_f32_16x16x4_f32(false, a, false, b0,
                                                           (short)0, acc[0], false, false);
            acc[1] = __builtin_amdgcn_wmma_f32_16x16x4_f32(false, a, false, b1,
                                                           (short)0, acc[1], false, false);
            acc[2] = __builtin_amdgcn_wmma_f32_16x16x4_f32(false, a, false, b2,
                                                           (short)0, acc[2], false, false);
            acc[3] = __builtin_amdgcn_wmma_f32_16x16x4_f32(false, a, false, b3,
                                                           (short)0, acc[3], false, false);
        }
    }

    // D layout: VGPR r, lanes 0-15 -> row r, lanes 16-31 -> row r+8; col = l16.
    const int ccol = which * 16 + l16;
#pragma unroll
    for (int b = 0; b < 4; ++b) {
#pragma unroll
        for (int r = 0; r < 8; ++r) {
            const int row = n0 + r + 8 * half;
            out[((size_t)b * NN + row) * 32 + ccol] = acc[b][r];
        }
    }
}

extern "C" void kernel_launch(void* const* d_in, const int* in_sizes, int n_in,
                              void* d_out, int out_size, void* d_ws, size_t ws_size,
                              hipStream_t stream) {
    const float* inputs = (const float*)d_in[0];
    const float* hidden = (const float*)d_in[1];
    const float* lap0   = (const float*)d_in[2];
    const float* lap1   = (const float*)d_in[3];
    const float* Wm     = (const float*)d_in[4];
    const float* bm     = (const float*)d_in[5];
    const float* Wv     = (const float*)d_in[6];
    const float* bv     = (const float*)d_in[7];
    const float* iwm    = (const float*)d_in[8];
    const float* iwv    = (const float*)d_in[9];
    float* out = (float*)d_out;

    float* M0 = (float*)d_ws;                    // N*64 floats = 2 MB
    float* M1 = M0 + (size_t)NN * 64;            // N*64 floats = 2 MB

    rtgcn_stage1<<<(BB * NN) / 256, 256, 0, stream>>>(
        inputs, hidden, Wm, bm, Wv, bv, iwm, iwv, M0, M1);

    dim3 g2(NN / 16, 2);
    rtgcn_stage2<<<g2, 32, 0, stream>>>(lap0, lap1, M0, M1, out);
}